// QuaternionLinear_54159537602595
// MI455X (gfx1250) — compile-verified
//
#include <hip/hip_runtime.h>

typedef __attribute__((ext_vector_type(2))) float v2f;
typedef __attribute__((ext_vector_type(8))) float v8f;

#define M_TOTAL 32768
#define N_TOTAL 1024
#define K_TOTAL 1024
#define BM 128
#define BN 128
#define BK 16
#define NSTAGES (K_TOTAL / BK)      // 64
#define AS_STRIDE 20                 // padded: conflict-free frag reads, 16B-aligned rows
#define BS_STRIDE 144                // padded so half-wave banks are disjoint
#define AS_ELEMS (BM * AS_STRIDE)    // 2560 floats
#define BS_ELEMS (BK * BS_STRIDE)    // 2304 floats

// Quaternion structure: B[k=4*iq+ci][n=4*oq+co] = sign * W_{ci^co}[oq][iq]
// Negative entries at bit co*4+ci => 0x428E
#define QSIGN_MASK 0x428E

#if defined(__gfx1250__) && \
    __has_builtin(__builtin_amdgcn_global_load_async_to_lds_b128) && \
    __has_builtin(__builtin_amdgcn_s_wait_asynccnt)
#define USE_ASYNC_A 1
typedef __attribute__((__vector_size__(16))) int vi4;
typedef __attribute__((address_space(1))) vi4 gbl_vi4;   // global source
typedef __attribute__((address_space(3))) vi4 lds_vi4;   // LDS destination
#else
#define USE_ASYNC_A 0
#endif

__global__ __launch_bounds__(256) void quat_gemm_f32_wmma(
    const float* __restrict__ X,
    const float* __restrict__ Wr,
    const float* __restrict__ Wi,
    const float* __restrict__ Wj,
    const float* __restrict__ Wk,
    const float* __restrict__ bias,
    float* __restrict__ Out)
{
    __shared__ float As[2][AS_ELEMS];
    __shared__ float Bs[2][BS_ELEMS];

    const int tid  = threadIdx.x;
    const int lane = tid & 31;
    const int wave = tid >> 5;
    const int wm   = wave >> 2;   // 0..1 : 64 M-rows per wave
    const int wn   = wave & 3;    // 0..3 : 32 N-cols per wave
    const int half = lane >> 4;   // 0..1
    const int lm   = lane & 15;

    const int m0 = blockIdx.y * BM;
    const int n0 = blockIdx.x * BN;

    v8f acc[4][2] = {};

    // ---- A-tile load mapping: thread -> (row, 8-float segment)
    const int ar = tid >> 1;             // 0..127
    const int ac = (tid & 1) * 8;        // 0 or 8
    const size_t xbase = (size_t)(m0 + ar) * K_TOTAL + (size_t)ac;  // add k0 per stage
    const int aoff = ar * AS_STRIDE + ac;

    // ---- B-tile build mapping: thread -> 8 consecutive elements of one k-row
    const int bbase = tid * 8;           // 0..2040
    const int bkk   = bbase >> 7;        // k-row 0..15
    const int bnn   = bbase & 127;       // n offset, multiple of 8
    const int boff  = bkk * BS_STRIDE + bnn;

    // Gather next-stage B weights into registers (sign applied in VALU).
    auto gather_b = [&](int k0, float bw[8]) {
        const int k  = k0 + bkk;
        const int ci = k & 3;
        const int iq = k >> 2;
        #pragma unroll
        for (int j = 0; j < 8; ++j) {
            const int nn  = bnn + j;
            const int co  = j & 3;               // bnn is a multiple of 8
            const int oq  = (n0 + nn) >> 2;
            const int sel = ci ^ co;
            const float* wp = (sel == 0) ? Wr : (sel == 1) ? Wi
                            : (sel == 2) ? Wj : Wk;
            const float w   = wp[oq * 256 + iq];
            const int   neg = (QSIGN_MASK >> (co * 4 + ci)) & 1;
            bw[j] = neg ? -w : w;
        }
    };

    auto issue_a = [&](int k0, int buf, float4& pa0, float4& pa1) {
#if USE_ASYNC_A
        (void)pa0; (void)pa1;
        __builtin_amdgcn_global_load_async_to_lds_b128(
            (gbl_vi4*)&X[xbase + k0],
            (lds_vi4*)&As[buf][aoff], 0, 0);
        __builtin_amdgcn_global_load_async_to_lds_b128(
            (gbl_vi4*)&X[xbase + k0 + 4],
            (lds_vi4*)&As[buf][aoff + 4], 0, 0);
#else
        (void)buf;
        pa0 = *reinterpret_cast<const float4*>(&X[xbase + k0]);
        pa1 = *reinterpret_cast<const float4*>(&X[xbase + k0 + 4]);
#endif
    };

    // ================= prologue: stage 0 into buffer 0 =================
    {
        float4 pa0, pa1;
        float bw[8];
        issue_a(0, 0, pa0, pa1);
        gather_b(0, bw);
#if !USE_ASYNC_A
        *reinterpret_cast<float4*>(&As[0][aoff])     = pa0;
        *reinterpret_cast<float4*>(&As[0][aoff + 4]) = pa1;
#endif
        *reinterpret_cast<float4*>(&Bs[0][boff])     = *reinterpret_cast<float4*>(&bw[0]);
        *reinterpret_cast<float4*>(&Bs[0][boff + 4]) = *reinterpret_cast<float4*>(&bw[4]);
#if USE_ASYNC_A
        __builtin_amdgcn_s_wait_asynccnt(0);
#endif
        __syncthreads();
    }

    // ================= pipelined main loop =================
    for (int s = 0; s < NSTAGES; ++s) {
        const int cur = s & 1;
        const int nxt = cur ^ 1;
        const float* Asb = As[cur];
        const float* Bsb = Bs[cur];
        const bool has_next = (s + 1 < NSTAGES);

        // --- issue stage s+1 global traffic before the compute block
        float4 pa0, pa1;
        float bw[8];
        if (has_next) {
            issue_a((s + 1) * BK, nxt, pa0, pa1);
            gather_b((s + 1) * BK, bw);
        }

        // --- compute stage s: 4 K-steps of 4, 8 WMMA tiles each
        #pragma unroll
        for (int kk = 0; kk < BK; kk += 4) {
            const int kb = kk + half * 2;  // lanes 0-15: K={kk,kk+1}; 16-31: K={kk+2,kk+3}
            v2f a[4], b[2];
            #pragma unroll
            for (int mt = 0; mt < 4; ++mt) {
                const int m = wm * 64 + mt * 16 + lm;
                a[mt] = *reinterpret_cast<const v2f*>(&Asb[m * AS_STRIDE + kb]);
            }
            #pragma unroll
            for (int nt = 0; nt < 2; ++nt) {
                const int n = wn * 32 + nt * 16 + lm;
                b[nt].x = Bsb[kb * BS_STRIDE + n];
                b[nt].y = Bsb[(kb + 1) * BS_STRIDE + n];
            }
            #pragma unroll
            for (int mt = 0; mt < 4; ++mt) {
                #pragma unroll
                for (int nt = 0; nt < 2; ++nt) {
                    acc[mt][nt] = __builtin_amdgcn_wmma_f32_16x16x4_f32(
                        /*neg_a=*/false, a[mt],
                        /*neg_b=*/false, b[nt],
                        /*c_mod=*/(short)0, acc[mt][nt],
                        /*reuse_a=*/false, /*reuse_b=*/false);
                }
            }
        }

        // --- commit stage s+1 into the other buffer
        if (has_next) {
#if !USE_ASYNC_A
            *reinterpret_cast<float4*>(&As[nxt][aoff])     = pa0;
            *reinterpret_cast<float4*>(&As[nxt][aoff + 4]) = pa1;
#endif
            *reinterpret_cast<float4*>(&Bs[nxt][boff])     = *reinterpret_cast<float4*>(&bw[0]);
            *reinterpret_cast<float4*>(&Bs[nxt][boff + 4]) = *reinterpret_cast<float4*>(&bw[4]);
#if USE_ASYNC_A
            __builtin_amdgcn_s_wait_asynccnt(0);
#endif
        }
        __syncthreads();
    }

    // ================= epilogue: C VGPR v -> row v + 8*half, col lm =================
    #pragma unroll
    for (int nt = 0; nt < 2; ++nt) {
        const int n = n0 + wn * 32 + nt * 16 + lm;
        const float bv = bias[n];
        #pragma unroll
        for (int mt = 0; mt < 4; ++mt) {
            #pragma unroll
            for (int v = 0; v < 8; ++v) {
                const int m = m0 + wm * 64 + mt * 16 + half * 8 + v;
                Out[(size_t)m * N_TOTAL + n] = acc[mt][nt][v] + bv;
            }
        }
    }
}

extern "C" void kernel_launch(void* const* d_in, const int* in_sizes, int n_in,
                              void* d_out, int out_size, void* d_ws, size_t ws_size,
                              hipStream_t stream) {
    const float* X    = (const float*)d_in[0];
    const float* Wr   = (const float*)d_in[1];
    const float* Wi   = (const float*)d_in[2];
    const float* Wj   = (const float*)d_in[3];
    const float* Wk   = (const float*)d_in[4];
    const float* bias = (const float*)d_in[5];
    float* Out = (float*)d_out;

    dim3 grid(N_TOTAL / BN, M_TOTAL / BM);   // (8, 256)
    quat_gemm_f32_wmma<<<grid, 256, 0, stream>>>(X, Wr, Wi, Wj, Wk, bias, Out);
}